// CurveGraphic2d_13056700579901
// MI455X (gfx1250) — compile-verified
//
#include <hip/hip_runtime.h>
#include <math.h>

typedef __attribute__((ext_vector_type(2))) float v2f;
typedef __attribute__((ext_vector_type(8))) float v8f;

#define CANVAS_H 1024
#define CANVAS_W 1024
#define CURVE_WIDTH 8.0f
#define EPS 1e-6f
#define NUM_SAMPLES 64

// D = A(16x4) * B(4x16) + C via V_WMMA_F32_16X16X4_F32
static __device__ __forceinline__ v8f wmma_f32_k4(v2f a, v2f b, v8f c) {
    // 8 args: (neg_a, A, neg_b, B, c_mod, C, reuse_a, reuse_b)
    return __builtin_amdgcn_wmma_f32_16x16x4_f32(false, a, false, b, (short)0, c,
                                                 false, false);
}

// min over the two lane-halves: lanes L and L^16 hold the two M-halves of the
// same pixel column. ds_swizzle SWAPX16: offset[14:10]=xor=0x10, and=0x1f.
static __device__ __forceinline__ float half_swap_min(float m) {
    int o = __builtin_amdgcn_ds_swizzle(__float_as_int(m), 0x401F);
    return fminf(m, __int_as_float(o));
}

// min over all 32 D elements (4 tiles x 8 row-VGPRs) for this lane's column
static __device__ __forceinline__ float min32(v8f d0, v8f d1, v8f d2, v8f d3) {
    float m = fminf(fminf(d0[0], d1[0]), fminf(d2[0], d3[0]));
#pragma unroll
    for (int i = 1; i < 8; ++i)
        m = fminf(m, fminf(fminf(d0[i], d1[i]), fminf(d2[i], d3[i])));
    return m;
}

__global__ void __launch_bounds__(256)
curve_graphic_wmma_kernel(const float* __restrict__ cps, float* __restrict__ out) {
    const int lane        = threadIdx.x & 31;
    const int waveInBlock = threadIdx.x >> 5;
    const int waveId      = blockIdx.x * (blockDim.x >> 5) + waveInBlock;

    // Each wave renders 64 contiguous pixels of one row (64 | 1024 -> no row cross)
    const int   pixBase = waveId * 64;
    const float y       = (float)(pixBase >> 10);       // row index
    const int   xBase   = pixBase & (CANVAS_W - 1);

    // Uniform control points, scaled to pixel coords.
    // Reference: component 0 is y-like (scaled by H), component 1 is x-like (by W).
    const float p0y = cps[0] * (float)CANVAS_H, p0x = cps[1] * (float)CANVAS_W;
    const float p1y = cps[2] * (float)CANVAS_H, p1x = cps[3] * (float)CANVAS_W;
    const float p2y = cps[4] * (float)CANVAS_H, p2x = cps[5] * (float)CANVAS_W;
    const float p3y = cps[6] * (float)CANVAS_H, p3x = cps[7] * (float)CANVAS_W;

    const int  mrow   = lane & 15;
    const bool hiHalf = lane >= 16;

    // ---- Build 4 A tiles (samples on M axis). Row m of tile t = sample 16t+m:
    //   K0..K3 = (-2*S0, -2*S1, |S|^2, 0)
    // A layout (f32 16x4): lanes 0-15 carry (K0,K1) of row M=lane,
    //                      lanes 16-31 carry (K2,K3) of row M=lane-16.
    v2f A[4];
#pragma unroll
    for (int t = 0; t < 4; ++t) {
        const int   sidx = t * 16 + mrow;
        const float tt   = (float)sidx * (1.0f / (float)(NUM_SAMPLES - 1));
        const float u    = 1.0f - tt;
        const float b0   = u * u * u;
        const float b1   = 3.0f * tt * u * u;
        const float b2   = 3.0f * tt * tt * u;
        const float b3   = tt * tt * tt;
        const float S0   = b0 * p0y + b1 * p1y + b2 * p2y + b3 * p3y;
        const float S1   = b0 * p0x + b1 * p1x + b2 * p2x + b3 * p3x;
        // per-lane selects (v_cndmask), no EXEC divergence around WMMA
        A[t].x = hiHalf ? (S0 * S0 + S1 * S1) : (-2.0f * S0);
        A[t].y = hiHalf ? 0.0f               : (-2.0f * S1);
    }

    // ---- Four pixel groups of 16 (pixels on N axis). B col n = (y, x_n, 1, 0)
    // After half_swap_min, every lane holds the full 64-sample min for pixel
    // (lane & 15) of that group.
    float rmin[4];
#pragma unroll
    for (int g = 0; g < 4; ++g) {
        v2f B;
        B.x = hiHalf ? 1.0f : y;                                  // K0 / K2
        B.y = hiHalf ? 0.0f : (float)(xBase + g * 16 + mrow);     // K1 / K3

        v8f c  = {};
        v8f d0 = wmma_f32_k4(A[0], B, c);
        v8f d1 = wmma_f32_k4(A[1], B, c);
        v8f d2 = wmma_f32_k4(A[2], B, c);
        v8f d3 = wmma_f32_k4(A[3], B, c);

        rmin[g] = half_swap_min(min32(d0, d1, d2, d3));
    }

    // Lane L owns pixels xBase+L (groups 0/1) and xBase+32+L (groups 2/3).
    // Branch-free selects (constant indices -> no scratch spills).
    const float mA = hiHalf ? rmin[1] : rmin[0];
    const float mB = hiHalf ? rmin[3] : rmin[2];

    const float pxA = (float)(xBase + lane);
    const float pxB = pxA + 32.0f;
    const float yy  = y * y;

    float dA = sqrtf(fmaxf(mA + (yy + pxA * pxA), 0.0f));   // + |p|^2
    float dB = sqrtf(fmaxf(mB + (yy + pxB * pxB), 0.0f));

    const float vA = 1.0f - (dA * (1.0f / CURVE_WIDTH) + EPS);  // AA_FACTOR==1
    const float vB = 1.0f - (dB * (1.0f / CURVE_WIDTH) + EPS);

    out[pixBase + lane]      = (dA < CURVE_WIDTH) ? vA : 0.0f;
    out[pixBase + 32 + lane] = (dB < CURVE_WIDTH) ? vB : 0.0f;
}

extern "C" void kernel_launch(void* const* d_in, const int* in_sizes, int n_in,
                              void* d_out, int out_size, void* d_ws, size_t ws_size,
                              hipStream_t stream) {
    (void)in_sizes; (void)n_in; (void)out_size; (void)d_ws; (void)ws_size;
    const float* cps = (const float*)d_in[0];   // 4 control points x 2 (f32)
    float*       out = (float*)d_out;           // 1024*1024 f32

    // 1024*1024 px / 64 px-per-wave = 16384 waves; 8 waves (256 thr) per block
    const int totalWaves    = (CANVAS_H * CANVAS_W) / 64;
    const int wavesPerBlock = 8;
    dim3 grid(totalWaves / wavesPerBlock);
    dim3 block(wavesPerBlock * 32);
    hipLaunchKernelGGL(curve_graphic_wmma_kernel, grid, block, 0, stream, cps, out);
}